// StericClashConstraint_61048665145613
// MI455X (gfx1250) — compile-verified
//
#include <hip/hip_runtime.h>

typedef __attribute__((ext_vector_type(2))) float v2f;
typedef __attribute__((ext_vector_type(8))) float v8f;

#define MIN_DISTANCE 1.0f
#define CONSTRAINT_WEIGHT 0.02f
#define CHUNK_PTS 2048   // points staged per LDS chunk: 2048*12B = 24 KB

// Pass-through copy of pos into d_out, and zero the trailing loss slot.
__global__ void steric_init_copy(const float* __restrict__ pos,
                                 float* __restrict__ out, int total) {
    int idx = blockIdx.x * blockDim.x + threadIdx.x;
    if (idx < total) out[idx] = pos[idx];
    if (idx == 0) out[total] = 0.0f;
}

// One wave32 per 16-row tile of the pairwise matrix. B tiles are staged
// through LDS with gfx1250 async global->LDS copies, gram dot-products via
// V_WMMA_F32_16X16X4_F32 (K = x,y,z,0), epilogue uses raw v_sqrt_f32.
__global__ __launch_bounds__(256) void steric_tile_kernel(
    const float* __restrict__ pos, float* __restrict__ loss,
    int nblk, float scale) {

    __shared__ float sp[CHUNK_PTS * 3];

    const int  lane   = threadIdx.x & 31;
    const int  wave   = blockIdx.x * (blockDim.x >> 5) + (threadIdx.x >> 5);
    const bool active = wave < nblk;

    const int  li      = lane & 15;     // point index within a 16-tile
    const bool hi      = lane >= 16;    // high half of wave
    const int  rowbase = hi ? 8 : 0;    // D rows for this half: v + rowbase

    // ---- A tile (rows wave*16 .. +15); K layout: lane<16 -> (x,y), lane>=16 -> (z,0)
    const int iw = active ? wave : 0;
    const float* pi = pos + (iw * 16 + li) * 3;
    const float ax = pi[0], ay = pi[1], az = pi[2];
    v2f A = { hi ? az : ax, hi ? 0.0f : ay };

    // full |p_i|^2 directly (no cross-lane op); lane l holds point (l&15)
    const float sqi = ax * ax + ay * ay + az * az;
    float sqi_row[8];
#pragma unroll
    for (int v = 0; v < 8; ++v)
        sqi_row[v] = __shfl(sqi, v + rowbase, 32);   // setup-only bpermute

    float acc = 0.0f;
    const int npts = nblk * 16;

    for (int base = 0; base < npts; base += CHUNK_PTS) {
        const int cpts = (npts - base < CHUNK_PTS) ? (npts - base) : CHUNK_PTS;
        const int ndw  = cpts * 3;                       // dwords to stage

        // ---- async stage pos[base .. base+cpts) into LDS, 16B per lane per op
        for (int t = threadIdx.x * 4; t < ndw; t += blockDim.x * 4) {
            unsigned ldsoff = (unsigned)(size_t)&sp[t];  // LDS byte offset
            const float* g  = pos + base * 3 + t;        // 16B-aligned
            asm volatile("global_load_async_to_lds_b128 %0, %1, off"
                         :: "v"(ldsoff), "v"(g) : "memory");
        }
        asm volatile("s_wait_asynccnt 0x0" ::: "memory");
        __syncthreads();

        if (active) {
            const int jtiles = cpts >> 4;
            for (int jt = 0; jt < jtiles; ++jt) {
                const int jb = (base >> 4) + jt;

                // B tile from LDS: same per-lane layout as A
                const float* q = &sp[(jt * 16 + li) * 3];
                const float bx = q[0], by = q[1], bz = q[2];
                v2f B = { hi ? bz : bx, hi ? 0.0f : by };
                const float sqj = bx * bx + by * by + bz * bz;

                v8f C = {};
                v8f D = __builtin_amdgcn_wmma_f32_16x16x4_f32(
                    /*neg_a=*/false, A, /*neg_b=*/false, B,
                    /*c_mod=*/(short)0, C, /*reuse_a=*/false, /*reuse_b=*/false);

                const bool dt = (jb == wave);
#pragma unroll
                for (int v = 0; v < 8; ++v) {
                    float d2 = sqi_row[v] + sqj - 2.0f * D[v];
                    d2 = fmaxf(d2, 0.0f);
                    // raw v_sqrt_f32: ~1 ulp, plenty for max(1-dist, 0)
                    float viol = fmaxf(MIN_DISTANCE - __builtin_amdgcn_sqrtf(d2), 0.0f);
                    if (dt && (li == v + rowbase)) viol = 0.0f;  // mask diagonal
                    acc += viol;
                }
            }
        }
        __syncthreads();
    }

    // wave32 reduction, then one atomic per active wave
#pragma unroll
    for (int off = 16; off > 0; off >>= 1)
        acc += __shfl_xor(acc, off, 32);
    if (active && lane == 0) atomicAdd(loss, acc * scale);
}

extern "C" void kernel_launch(void* const* d_in, const int* in_sizes, int n_in,
                              void* d_out, int out_size, void* d_ws, size_t ws_size,
                              hipStream_t stream) {
    const float* pos = (const float*)d_in[0];
    float*       out = (float*)d_out;

    const int total = in_sizes[0];      // N*3 floats
    const int N     = total / 3;        // 16384
    const int nblk  = N / 16;           // 1024 row/col tiles
    const float scale = CONSTRAINT_WEIGHT / ((float)N * (float)N);

    // 1) copy pos -> out, zero the loss accumulator at out[N*3]
    steric_init_copy<<<(total + 255) / 256, 256, 0, stream>>>(pos, out, total);

    // 2) all-pairs tiles: one wave per 16-row block, 8 waves per 256-thread block
    const int wavesPerBlock = 8;
    const int blocks = (nblk + wavesPerBlock - 1) / wavesPerBlock;
    steric_tile_kernel<<<blocks, wavesPerBlock * 32, 0, stream>>>(
        pos, out + total, nblk, scale);
}